// Encoder_3496103379229
// MI455X (gfx1250) — compile-verified
//
#include <hip/hip_runtime.h>
#include <hip/hip_bf16.h>

// Problem constants (from reference)
#define BB 8
#define SS 4096
#define NN 2048
#define HH 768
#define KK 50

typedef __attribute__((ext_vector_type(2))) float v2f;
typedef __attribute__((ext_vector_type(8))) float v8f;

// ---------------------------------------------------------------------------
// Phase 1: per-token anchor scores.
//   tok4[row, n] = seq[row, :] . anchor[n, :]   for n<3 (cols 3..15 are zero pad)
// Skinny fp32 GEMM (32768 x 768) x (768 x 16) via V_WMMA_F32_16X16X4_F32.
// One wave = one 16-row M tile; 8 waves/block; anchor^T (768x16) staged in LDS.
// This is the single streaming pass over the 100 MB tensor (bandwidth bound).
// ---------------------------------------------------------------------------
__global__ __launch_bounds__(256) void k_tok_scores(const float* __restrict__ seq,
                                                    const float* __restrict__ anchor,
                                                    float* __restrict__ tok4) {
  __shared__ float ldsB[HH * 16];  // B[k][n] = anchor[n][k], zero-padded n>=3 (48 KB)
  const int tid = threadIdx.x;
  for (int i = tid; i < HH * 16; i += 256) {
    const int k = i >> 4, n = i & 15;
    ldsB[i] = (n < 3) ? anchor[n * HH + k] : 0.0f;
  }
  __syncthreads();

  const int wave = tid >> 5, lane = tid & 31;
  const int lrow = lane & 15;          // A row / B,D column within tile
  const int lk   = (lane >> 4) << 1;   // K sub-offset: 0 or 2
  const long r0  = (long)blockIdx.x * 128 + wave * 16;
  const float* arow = seq + (r0 + lrow) * HH;

  v8f acc = {};
  for (int k0 = 0; k0 < HH; k0 += 4) {
    v2f a = *(const v2f*)(arow + k0 + lk);            // A[row, k0+lk .. +1]
    v2f b;
    b.x = ldsB[(k0 + lk) * 16 + lrow];                // B[k0+lk,   col]
    b.y = ldsB[(k0 + lk + 1) * 16 + lrow];            // B[k0+lk+1, col]
    acc = __builtin_amdgcn_wmma_f32_16x16x4_f32(false, a, false, b,
                                                (short)0, acc, false, false);
  }
  // D layout: vgpr i -> row i (lanes 0-15) / row i+8 (lanes 16-31), col = lane&15
  if (lrow < 4) {
    const int rb = (lane >> 4) << 3;
    for (int i = 0; i < 8; ++i) tok4[(r0 + rb + i) * 4 + lrow] = acc[i];
  }
}

// ---------------------------------------------------------------------------
// Phase 2: span score = max_c ( mean over span of token scores[c] )
// ---------------------------------------------------------------------------
__global__ void k_span_scores(const float* __restrict__ tok4,
                              const int* __restrict__ starts,
                              const int* __restrict__ lens,
                              float* __restrict__ scoremax) {
  const int i = blockIdx.x * blockDim.x + threadIdx.x;
  if (i >= BB * NN) return;
  const int b = i / NN;
  const int s = starts[i];
  const int nt = lens[i] + 1;  // 1..8
  const float* p = tok4 + (long)(b * SS + s) * 4;
  float c0 = 0.f, c1 = 0.f, c2 = 0.f;
  for (int t = 0; t < nt; ++t) {
    c0 += p[t * 4 + 0];
    c1 += p[t * 4 + 1];
    c2 += p[t * 4 + 2];
  }
  scoremax[i] = fmaxf(c0, fmaxf(c1, c2)) / (float)nt;
}

// ---------------------------------------------------------------------------
// Phase 3: top-50 per batch, jax.lax.top_k order (desc value, ties -> low idx).
// One block per batch; 50 iterative argmax reductions over 2048 values in LDS.
// ---------------------------------------------------------------------------
__global__ __launch_bounds__(256) void k_topk(const float* __restrict__ scoremax,
                                              int* __restrict__ topidx) {
  __shared__ float vals[NN];
  __shared__ float rv[256];
  __shared__ int ri[256];
  const int b = blockIdx.x, tid = threadIdx.x;
  for (int j = tid; j < NN; j += 256) vals[j] = scoremax[b * NN + j];
  __syncthreads();
  for (int kk = 0; kk < KK; ++kk) {
    float bv = -INFINITY;
    int bi = NN;
    for (int j = tid; j < NN; j += 256) {
      const float v = vals[j];
      if (v > bv || (v == bv && j < bi)) { bv = v; bi = j; }
    }
    rv[tid] = bv; ri[tid] = bi;
    __syncthreads();
    for (int off = 128; off > 0; off >>= 1) {
      if (tid < off) {
        const float v = rv[tid + off];
        const int j2 = ri[tid + off];
        if (v > rv[tid] || (v == rv[tid] && j2 < ri[tid])) { rv[tid] = v; ri[tid] = j2; }
      }
      __syncthreads();
    }
    if (tid == 0) { topidx[b * KK + kk] = ri[0]; vals[ri[0]] = -INFINITY; }
    __syncthreads();
  }
}

// ---------------------------------------------------------------------------
// Phase 4: gather mean embeddings of selected spans -> sel (400 x 768)
// ---------------------------------------------------------------------------
__global__ void k_sel(const float* __restrict__ seq,
                      const int* __restrict__ starts,
                      const int* __restrict__ lens,
                      const int* __restrict__ topidx,
                      float* __restrict__ sel) {
  const int blk = blockIdx.x;          // b*KK + k
  const int b = blk / KK;
  const int idx = topidx[blk];
  const int s = starts[b * NN + idx];
  const int nt = lens[b * NN + idx] + 1;
  const float* base = seq + (long)(b * SS + s) * HH;
  const float inv = 1.0f / (float)nt;
  for (int h = threadIdx.x; h < HH; h += blockDim.x) {
    float a = 0.f;
    for (int t = 0; t < nt; ++t) a += base[t * HH + h];
    sel[(long)blk * HH + h] = a * inv;
  }
}

// ---------------------------------------------------------------------------
// Phase 5: projections  proj (400 x 48) = sel (400 x 768) . Wall^T (768 x 48)
//   Wall rows: [0..3]=Wr_h  [4..7]=Wr_t  [8..27]=Wn_h  [28..47]=Wn_t
// fp32 WMMA 16x16x4; 25 M-tiles x 3 N-tiles; 4 waves/block share a 48 KB
// LDS-staged B tile. grid = (7, 3), block = 128.
// ---------------------------------------------------------------------------
__global__ __launch_bounds__(128) void k_proj(const float* __restrict__ sel,
                                              const float* __restrict__ rel_emb,
                                              const float* __restrict__ nota_emb,
                                              float* __restrict__ proj) {
  __shared__ float ldsB[HH * 16];
  const int tid = threadIdx.x;
  const int ntile = blockIdx.y;
  for (int i = tid; i < HH * 16; i += 128) {
    const int k = i >> 4, nn = i & 15;
    const int j = ntile * 16 + nn;   // 0..47
    float v;
    if (j < 4)       v = rel_emb[j * (2 * HH) + k];
    else if (j < 8)  v = rel_emb[(j - 4) * (2 * HH) + HH + k];
    else if (j < 28) v = nota_emb[(j - 8) * (2 * HH) + k];
    else             v = nota_emb[(j - 28) * (2 * HH) + HH + k];
    ldsB[i] = v;
  }
  __syncthreads();

  const int wave = tid >> 5, lane = tid & 31;
  const int mtile = blockIdx.x * 4 + wave;
  if (mtile >= 25) return;  // wave-uniform, no barriers after this point
  const int lrow = lane & 15;
  const int lk = (lane >> 4) << 1;
  const float* arow = sel + (long)(mtile * 16 + lrow) * HH;

  v8f acc = {};
  for (int k0 = 0; k0 < HH; k0 += 4) {
    v2f a = *(const v2f*)(arow + k0 + lk);
    v2f b;
    b.x = ldsB[(k0 + lk) * 16 + lrow];
    b.y = ldsB[(k0 + lk + 1) * 16 + lrow];
    acc = __builtin_amdgcn_wmma_f32_16x16x4_f32(false, a, false, b,
                                                (short)0, acc, false, false);
  }
  const int rb = (lane >> 4) << 3;
  for (int i = 0; i < 8; ++i) {
    const int m = mtile * 16 + rb + i;
    proj[m * 48 + ntile * 16 + lrow] = acc[i];
  }
}

// ---------------------------------------------------------------------------
// Phase 6: out[b,k1,k2] = [ max_m(nh[k1,m]+nt[k2,m]), hh[k1,r]+ht[k2,r] (r<4) ]
// ---------------------------------------------------------------------------
__global__ void k_final(const float* __restrict__ proj, float* __restrict__ out) {
  const int i = blockIdx.x * blockDim.x + threadIdx.x;
  if (i >= BB * KK * KK) return;
  const int k2 = i % KK;
  const int t = i / KK;
  const int k1 = t % KK;
  const int b = t / KK;
  const float* p1 = proj + (b * KK + k1) * 48;
  const float* p2 = proj + (b * KK + k2) * 48;
  float nmax = -INFINITY;
  for (int m = 0; m < 20; ++m) nmax = fmaxf(nmax, p1[8 + m] + p2[28 + m]);
  float* o = out + (long)i * 5;
  o[0] = nmax;
  for (int r = 0; r < 4; ++r) o[1 + r] = p1[r] + p2[4 + r];
}

extern "C" void kernel_launch(void* const* d_in, const int* in_sizes, int n_in,
                              void* d_out, int out_size, void* d_ws, size_t ws_size,
                              hipStream_t stream) {
  const float* seq      = (const float*)d_in[0];  // (8,4096,768) f32
  const int*   starts   = (const int*)d_in[1];    // (8,2048) i32
  const int*   lens     = (const int*)d_in[2];    // (8,2048) i32
  const float* anchor   = (const float*)d_in[3];  // (3,768) f32
  const float* rel_emb  = (const float*)d_in[4];  // (4,1536) f32
  const float* nota_emb = (const float*)d_in[5];  // (20,1536) f32
  float* out = (float*)d_out;                     // (8,50,50,5) f32

  // Workspace layout (bytes), ~1.9 MB total
  char* ws = (char*)d_ws;
  float* tok4     = (float*)(ws + 0);        // 32768*4 f32   = 512 KB
  float* scoremax = (float*)(ws + 524288);   // 16384 f32     =  64 KB
  int*   topidx   = (int*)  (ws + 589824);   // 400 i32
  float* sel      = (float*)(ws + 591872);   // 400*768 f32   = 1.2 MB
  float* proj     = (float*)(ws + 1820672);  // 400*48 f32    =  75 KB

  k_tok_scores<<<256, 256, 0, stream>>>(seq, anchor, tok4);
  k_span_scores<<<(BB * NN + 255) / 256, 256, 0, stream>>>(tok4, starts, lens, scoremax);
  k_topk<<<BB, 256, 0, stream>>>(scoremax, topidx);
  k_sel<<<BB * KK, 256, 0, stream>>>(seq, starts, lens, topidx, sel);
  k_proj<<<dim3(7, 3), 128, 0, stream>>>(sel, rel_emb, nota_emb, proj);
  k_final<<<(BB * KK * KK + 255) / 256, 256, 0, stream>>>(proj, out);
}